// ssim_module_21242908246556
// MI455X (gfx1250) — compile-verified
//
#include <hip/hip_runtime.h>

// SSIM (window=3, stride=3, pad=1) on 16x3x512x512 f32 pairs -> scalar mean.
// Memory-bound (~100 MB @ 23.3 TB/s ≈ 4.3 us floor). Strategy: async
// global->LDS B128 staging (CDNA5 ASYNCcnt path), disjoint 3x3 weighted
// pooling from LDS, deterministic two-stage reduction.

#define IH 512
#define IW 512
#define OH 171
#define OW 171
#define NPLANES 48                 // 16 * 3
#define NBLOCKS (NPLANES * OH)     // 8208
#define TPB 256
#define SSIM_C1 1.0e-4f
#define SSIM_C2 9.0e-4f
#define N_OUT_VALS 1403568.0       // 16*3*171*171

// Async global->LDS builtin wants (global int4*, local int4*, imm, imm):
// compiler said param0 is 'int __attribute__((vector_size(16))) __device__ *'.
typedef int v4i __attribute__((ext_vector_type(4)));
typedef __attribute__((address_space(1))) v4i g_v4i;   // global ("device")
typedef __attribute__((address_space(3))) v4i l_v4i;   // LDS ("shared")

#if defined(__gfx1250__) && __has_builtin(__builtin_amdgcn_global_load_async_to_lds_b128)
#define USE_ASYNC_LDS 1
#else
#define USE_ASYNC_LDS 0
#endif

__global__ __launch_bounds__(TPB) void ssim_row_kernel(
    const float* __restrict__ img1, const float* __restrict__ img2,
    const float* __restrict__ win, float* __restrict__ partials)
{
    __shared__ float lds[6 * IW];   // rows 0..2: img1, rows 3..5: img2 (12 KB)
    __shared__ float red[TPB];

    const int bid   = blockIdx.x;
    const int plane = bid / OH;          // n*3 + c
    const int oy    = bid - plane * OH;
    const int tid   = threadIdx.x;
    const size_t pbase = (size_t)plane * (size_t)(IH * IW);
    const int iy0 = 3 * oy - 1;

    // Stage 6 rows (3 per image) = 768 float4 transfers; 3 per thread.
    for (int idx = tid; idx < 6 * (IW / 4); idx += TPB) {
        const int r     = idx >> 7;       // 0..5 (IW/4 == 128)
        const int chunk = idx & 127;      // float4 index within row
        const int dy    = (r < 3) ? r : r - 3;
        const int iy    = iy0 + dy;
        float* dstf = &lds[r * IW + chunk * 4];
        if (iy >= 0 && iy < IH) {
            const float* base = (r < 3) ? img1 : img2;
            const float* src  = base + pbase + (size_t)iy * IW + (size_t)chunk * 4;
#if USE_ASYNC_LDS
            __builtin_amdgcn_global_load_async_to_lds_b128(
                (g_v4i*)(uintptr_t)src, (l_v4i*)(uintptr_t)dstf,
                /*offset=*/0, /*cpol=*/0);
#else
            *(float4*)dstf = *(const float4*)src;
#endif
        } else {
            *(float4*)dstf = make_float4(0.f, 0.f, 0.f, 0.f);  // zero padding row
        }
    }
#if USE_ASYNC_LDS
#if __has_builtin(__builtin_amdgcn_s_wait_asynccnt)
    __builtin_amdgcn_s_wait_asynccnt(0);
#else
    asm volatile("s_wait_asynccnt 0" ::: "memory");
#endif
#endif
    __syncthreads();

    // 3x3 Gaussian window (identical across channels): first 9 floats.
    float wv[9];
#pragma unroll
    for (int k = 0; k < 9; ++k) wv[k] = win[k];

    float val = 0.f;
    if (tid < OW) {
        const int ix0 = 3 * tid - 1;     // ix0 == -1 only for tid == 0
        float s1 = 0.f, s2 = 0.f, s11 = 0.f, s22 = 0.f, s12 = 0.f;
#pragma unroll
        for (int dy = 0; dy < 3; ++dy) {
            const float* r1 = &lds[dy * IW];
            const float* r2 = &lds[(dy + 3) * IW];
#pragma unroll
            for (int dx = 0; dx < 3; ++dx) {
                const int ix  = ix0 + dx;
                const float w = wv[dy * 3 + dx];
                const float a = (ix >= 0) ? r1[ix] : 0.f;  // left zero-pad
                const float b = (ix >= 0) ? r2[ix] : 0.f;
                s1  += w * a;
                s2  += w * b;
                s11 += w * a * a;
                s22 += w * b * b;
                s12 += w * a * b;
            }
        }
        const float mu1s  = s1 * s1;
        const float mu2s  = s2 * s2;
        const float mu12  = s1 * s2;
        const float sig1  = s11 - mu1s;
        const float sig2  = s22 - mu2s;
        const float sig12 = s12 - mu12;
        val = ((2.f * mu12 + SSIM_C1) * (2.f * sig12 + SSIM_C2)) /
              ((mu1s + mu2s + SSIM_C1) * (sig1 + sig2 + SSIM_C2));
    }

    // Deterministic block tree reduction (256 = 2^8).
    red[tid] = val;
    __syncthreads();
#pragma unroll
    for (int s = TPB / 2; s > 0; s >>= 1) {
        if (tid < s) red[tid] += red[tid + s];
        __syncthreads();
    }
    if (tid == 0) partials[bid] = red[0];
}

__global__ __launch_bounds__(256) void ssim_reduce_kernel(
    const float* __restrict__ partials, float* __restrict__ out, int n)
{
    __shared__ double red[256];
    double acc = 0.0;
    for (int i = threadIdx.x; i < n; i += 256) acc += (double)partials[i];
    red[threadIdx.x] = acc;
    __syncthreads();
#pragma unroll
    for (int s = 128; s > 0; s >>= 1) {
        if (threadIdx.x < s) red[threadIdx.x] += red[threadIdx.x + s];
        __syncthreads();
    }
    if (threadIdx.x == 0) out[0] = (float)(red[0] / N_OUT_VALS);
}

extern "C" void kernel_launch(void* const* d_in, const int* in_sizes, int n_in,
                              void* d_out, int out_size, void* d_ws, size_t ws_size,
                              hipStream_t stream) {
    (void)in_sizes; (void)n_in; (void)out_size; (void)ws_size;
    const float* img1 = (const float*)d_in[0];   // [16,3,512,512] f32
    const float* img2 = (const float*)d_in[1];   // [16,3,512,512] f32
    const float* win  = (const float*)d_in[2];   // [3,1,3,3] f32 (all channels equal)
    float* partials = (float*)d_ws;              // NBLOCKS floats (~32.8 KB)

    ssim_row_kernel<<<NBLOCKS, TPB, 0, stream>>>(img1, img2, win, partials);
    ssim_reduce_kernel<<<1, 256, 0, stream>>>(partials, (float*)d_out, NBLOCKS);
}